// MSSSIM_33767032881849
// MI455X (gfx1250) — compile-verified
//
#include <hip/hip_runtime.h>
#include <hip/hip_bf16.h>
#include <math.h>

// ---------------------------------------------------------------------------
// MS-SSIM, fused per level for MI455X (gfx1250, wave32).
// Separable 11-tap Gaussian conv mapped onto V_WMMA_F32_16X16X4_F32 chains:
//   htmp(16..26 x 64) = in(.. x 76) * Gh   (banded 4x16 weight chunks as B)
//   out (16 x 64)     = Gv * htmp          (banded 16x4 weight chunks as A)
// The banded per-lane constants are identical for both passes.
// Tile loads go through GLOBAL_LOAD_ASYNC_TO_LDS_B128 (ASYNCcnt).
// ---------------------------------------------------------------------------

typedef __attribute__((ext_vector_type(2))) float v2f;
typedef __attribute__((ext_vector_type(8))) float v8f;
typedef __attribute__((ext_vector_type(4))) int v4i;

typedef __attribute__((address_space(1))) v4i* global_v4i_ptr;
typedef __attribute__((address_space(3))) v4i* lds_v4i_ptr;

#define NIMG 48            // 16 batch * 3 channels
#define C1F  4.0e-4f       // (0.01*2)^2
#define C2F  3.6e-3f       // (0.03*2)^2

// Gaussian(sigma=1.5, ws=11), normalized.
__constant__ float GW[11] = {
    0.00102838f, 0.00759876f, 0.03600077f, 0.10936069f, 0.21300553f,
    0.26601172f, 0.21300553f, 0.10936069f, 0.03600077f, 0.00759876f,
    0.00102838f};

__device__ inline v8f wmma_f32_16x16x4(v2f a, v2f b, v8f c) {
  // 8 args: (neg_a, A, neg_b, B, c_mod, C, reuse_a, reuse_b)
  return __builtin_amdgcn_wmma_f32_16x16x4_f32(false, a, false, b, (short)0, c,
                                               false, false);
}

#if defined(__gfx1250__) && \
    __has_builtin(__builtin_amdgcn_global_load_async_to_lds_b128)
#define HAVE_ASYNC_LDS 1
#endif

__device__ inline void wait_async_lds() {
#if defined(HAVE_ASYNC_LDS)
#if __has_builtin(__builtin_amdgcn_s_wait_asynccnt)
  __builtin_amdgcn_s_wait_asynccnt(0);
#else
  asm volatile("s_wait_asynccnt 0" ::: "memory");
#endif
#endif
}

__global__ __launch_bounds__(128, 1) void ssim_level_kernel(
    const float* __restrict__ img1, const float* __restrict__ img2, int H,
    int W, float* __restrict__ sacc, float* __restrict__ cacc) {
  __shared__ __attribute__((aligned(16))) float li1[26 * 76];
  __shared__ __attribute__((aligned(16))) float li2[26 * 76];
  __shared__ __attribute__((aligned(16))) float ht[5][28 * 65];
  __shared__ float rbuf[8];

  const int tid = threadIdx.x;
  const int lane = tid & 31;
  const int wv = tid >> 5;       // wave id 0..3
  const int half = lane >> 4;    // 0: lanes 0-15, 1: lanes 16-31
  const int mn = lane & 15;      // M (A) or N (B) index for this lane
  const int xw = wv << 4;        // 16-col slice owned by this wave

  const int r0 = blockIdx.y * 16;
  const int c0 = blockIdx.x * 64;
  const int img = blockIdx.z;
  const int oh = H - 10, ow = W - 10;

  const float* b1 = img1 + (size_t)img * H * W;
  const float* b2 = img2 + (size_t)img * H * W;

  // ---- cooperative tile load: 26 rows x 76 cols (19 x b128 per row) ----
  for (int idx = tid; idx < 26 * 19; idx += 128) {
    int rr = idx / 19;
    int c4 = (idx - rr * 19) * 4;
    int gr = r0 + rr;
    int gc = c0 + c4;
    float* l1 = &li1[rr * 76 + c4];
    float* l2 = &li2[rr * 76 + c4];
    if (gr < H && gc < W) {  // W%4==0, c4%4==0 -> whole group in-range
#if defined(HAVE_ASYNC_LDS)
      __builtin_amdgcn_global_load_async_to_lds_b128(
          (global_v4i_ptr)(b1 + (size_t)gr * W + gc), (lds_v4i_ptr)l1, 0, 0);
      __builtin_amdgcn_global_load_async_to_lds_b128(
          (global_v4i_ptr)(b2 + (size_t)gr * W + gc), (lds_v4i_ptr)l2, 0, 0);
#else
      float4 v1 = *(const float4*)(b1 + (size_t)gr * W + gc);
      float4 v2 = *(const float4*)(b2 + (size_t)gr * W + gc);
      *(float4*)l1 = v1;
      *(float4*)l2 = v2;
#endif
    } else {
      float4 z = make_float4(0.f, 0.f, 0.f, 0.f);
      *(float4*)l1 = z;
      *(float4*)l2 = z;
    }
  }
  wait_async_lds();
  __syncthreads();

  // ---- per-lane banded Gaussian constants (shared by H and V passes) ----
  // chunk c, lane: value = g[4c + {0,1|2,3} - mn]  (0 outside [0,10])
  float wx[7], wy[7];
#pragma unroll
  for (int c = 0; c < 7; ++c) {
    int j = 4 * c + 2 * half - mn;
    wx[c] = (j >= 0 && j < 11) ? GW[j] : 0.0f;
    int j1 = j + 1;
    wy[c] = (j1 >= 0 && j1 < 11) ? GW[j1] : 0.0f;
  }

  // ---- horizontal pass: htmp = in * Gh  (A = input rows, B = banded w) ----
#pragma unroll
  for (int t = 0; t < 2; ++t) {  // M-tiles: rows 0-15, 16-31 (26..31 junk)
    v8f h0 = {}, h1 = {}, h2 = {}, h3 = {}, h4 = {};
    int mrow = t * 16 + mn;
    if (mrow > 25) mrow = 25;  // clamp; junk rows masked later
#pragma unroll
    for (int c = 0; c < 7; ++c) {
      int p = xw + 4 * c + 2 * half;  // input col (max 75 < 76)
      float a1x = li1[mrow * 76 + p], a1y = li1[mrow * 76 + p + 1];
      float a2x = li2[mrow * 76 + p], a2y = li2[mrow * 76 + p + 1];
      v2f wvv;  wvv.x = wx[c];  wvv.y = wy[c];
      v2f A;
      A.x = a1x;        A.y = a1y;        h0 = wmma_f32_16x16x4(A, wvv, h0);
      A.x = a2x;        A.y = a2y;        h1 = wmma_f32_16x16x4(A, wvv, h1);
      A.x = a1x * a1x;  A.y = a1y * a1y;  h2 = wmma_f32_16x16x4(A, wvv, h2);
      A.x = a2x * a2x;  A.y = a2y * a2y;  h3 = wmma_f32_16x16x4(A, wvv, h3);
      A.x = a1x * a2x;  A.y = a1y * a2y;  h4 = wmma_f32_16x16x4(A, wvv, h4);
    }
#pragma unroll
    for (int i = 0; i < 8; ++i) {
      int r = t * 16 + i + 8 * half;
      if (r < 28) {  // rows 26,27 hold finite junk; zero-weighted in V pass
        int o = r * 65 + xw + mn;
        ht[0][o] = h0[i];
        ht[1][o] = h1[i];
        ht[2][o] = h2[i];
        ht[3][o] = h3[i];
        ht[4][o] = h4[i];
      }
    }
  }
  __syncthreads();

  // ---- vertical pass: out = Gv * htmp  (A = banded w, B = htmp rows) ----
  v8f m1 = {}, m2 = {}, e11 = {}, e22 = {}, e12 = {};
  const int x = xw + mn;
#pragma unroll
  for (int c = 0; c < 7; ++c) {
    int r = 4 * c + 2 * half;  // htmp row (max 27 < 28)
    v2f wvv;  wvv.x = wx[c];  wvv.y = wy[c];
    v2f B;
    B.x = ht[0][r * 65 + x]; B.y = ht[0][(r + 1) * 65 + x];
    m1 = wmma_f32_16x16x4(wvv, B, m1);
    B.x = ht[1][r * 65 + x]; B.y = ht[1][(r + 1) * 65 + x];
    m2 = wmma_f32_16x16x4(wvv, B, m2);
    B.x = ht[2][r * 65 + x]; B.y = ht[2][(r + 1) * 65 + x];
    e11 = wmma_f32_16x16x4(wvv, B, e11);
    B.x = ht[3][r * 65 + x]; B.y = ht[3][(r + 1) * 65 + x];
    e22 = wmma_f32_16x16x4(wvv, B, e22);
    B.x = ht[4][r * 65 + x]; B.y = ht[4][(r + 1) * 65 + x];
    e12 = wmma_f32_16x16x4(wvv, B, e12);
  }

  // ---- per-element SSIM / CS with boundary mask, local accumulate ----
  float ls = 0.f, lc = 0.f;
  const int gx = c0 + x;
#pragma unroll
  for (int i = 0; i < 8; ++i) {
    int gy = r0 + i + 8 * half;
    float mu1 = m1[i], mu2 = m2[i];
    float mu1s = mu1 * mu1, mu2s = mu2 * mu2, mu12 = mu1 * mu2;
    float s1 = e11[i] - mu1s, s2 = e22[i] - mu2s, s12 = e12[i] - mu12;
    float v1 = 2.f * s12 + C2F;
    float v2 = s1 + s2 + C2F;
    float cs = v1 / v2;
    float ss = ((2.f * mu12 + C1F) * v1) / ((mu1s + mu2s + C1F) * v2);
    if (gy < oh && gx < ow) {
      ls += ss;
      lc += cs;
    }
  }

  // ---- wave shuffle reduce, block combine, global atomic ----
#pragma unroll
  for (int o = 16; o > 0; o >>= 1) {
    ls += __shfl_down(ls, o, 32);
    lc += __shfl_down(lc, o, 32);
  }
  if (lane == 0) {
    rbuf[wv] = ls;
    rbuf[4 + wv] = lc;
  }
  __syncthreads();
  if (tid == 0) {
    atomicAdd(sacc, rbuf[0] + rbuf[1] + rbuf[2] + rbuf[3]);
    atomicAdd(cacc, rbuf[4] + rbuf[5] + rbuf[6] + rbuf[7]);
  }
}

__global__ void down2_kernel(const float* __restrict__ src,
                             float* __restrict__ dst, int H, int W) {
  int h2 = H >> 1, w2 = W >> 1;
  int n = NIMG * h2 * w2;
  int id = blockIdx.x * blockDim.x + threadIdx.x;
  if (id >= n) return;
  int xx = id % w2;
  int t = id / w2;
  int yy = t % h2;
  int im = t / h2;
  const float* s = src + (size_t)im * H * W;
  float v = 0.25f * (s[(2 * yy) * W + 2 * xx] + s[(2 * yy) * W + 2 * xx + 1] +
                     s[(2 * yy + 1) * W + 2 * xx] +
                     s[(2 * yy + 1) * W + 2 * xx + 1]);
  dst[(size_t)im * h2 * w2 + yy * w2 + xx] = v;
}

__global__ void zero_acc_kernel(float* acc) {
  if (threadIdx.x < 16) acc[threadIdx.x] = 0.f;
}

__global__ void finalize_kernel(const float* __restrict__ acc,
                                float* __restrict__ out) {
  const float Wt[5] = {0.0448f, 0.2856f, 0.3001f, 0.2363f, 0.1333f};
  float prod = 1.f;
  float mssim4 = 0.f;
  for (int l = 0; l < 5; ++l) {
    int H = 512 >> l;
    int o = H - 10;
    float cnt = (float)NIMG * (float)o * (float)o;
    float ms = (acc[l] / cnt + 1.f) * 0.5f;
    float mc = (acc[5 + l] / cnt + 1.f) * 0.5f;
    if (l < 4)
      prod *= powf(mc, Wt[l]);
    else
      mssim4 = ms;
  }
  // prod(pow1[:-1] * pow2[-1]) = (prod mcs_l^w_l, l<4) * mssim_4^(4*w_4)
  prod *= powf(mssim4, 4.f * Wt[4]);
  out[0] = 1.f - prod;
}

extern "C" void kernel_launch(void* const* d_in, const int* in_sizes, int n_in,
                              void* d_out, int out_size, void* d_ws,
                              size_t ws_size, hipStream_t stream) {
  (void)in_sizes; (void)n_in; (void)out_size; (void)ws_size;
  const float* cur1 = (const float*)d_in[0];
  const float* cur2 = (const float*)d_in[1];
  float* acc = (float*)d_ws;   // 16 floats of accumulators
  size_t off = 64;             // 256 B into ws: level image buffers

  zero_acc_kernel<<<1, 32, 0, stream>>>(acc);

  for (int l = 0; l < 5; ++l) {
    int H = 512 >> l, W = H;
    int oh = H - 10, ow = W - 10;
    dim3 grid((ow + 63) / 64, (oh + 15) / 16, NIMG);
    ssim_level_kernel<<<grid, 128, 0, stream>>>(cur1, cur2, H, W, acc + l,
                                                acc + 5 + l);
    if (l < 4) {
      int h2 = H / 2, w2 = W / 2;
      size_t n = (size_t)NIMG * h2 * w2;
      float* n1 = (float*)d_ws + off;
      off += n;
      float* n2 = (float*)d_ws + off;
      off += n;
      int tot = (int)n;
      down2_kernel<<<(tot + 255) / 256, 256, 0, stream>>>(cur1, n1, H, W);
      down2_kernel<<<(tot + 255) / 256, 256, 0, stream>>>(cur2, n2, H, W);
      cur1 = n1;
      cur2 = n2;
    }
  }
  finalize_kernel<<<1, 1, 0, stream>>>(acc, (float*)d_out);
}